// ChamferDistance_9887014716187
// MI455X (gfx1250) — compile-verified
//
#include <hip/hip_runtime.h>
#include <hip/hip_bf16.h>

typedef __attribute__((ext_vector_type(2))) float v2f;
typedef __attribute__((ext_vector_type(8))) float v8f;

#define COL_SPLIT 8

// ---------------------------------------------------------------------------
// Init per-row / per-col min buffers to +inf (as uint bit pattern).
// ---------------------------------------------------------------------------
__global__ void cd_init_kernel(unsigned* __restrict__ bits, int n) {
  int i = blockIdx.x * blockDim.x + threadIdx.x;
  if (i < n) bits[i] = 0x7F800000u;  // +inf
}

// ---------------------------------------------------------------------------
// Each wave owns 16 rows of X and sweeps a segment of Y columns in 16-wide
// tiles. One V_WMMA_F32_16X16X4_F32 per tile produces the squared distances
// directly by exploiting all 4 K slots:
//   A[m] = (-2*x0, -2*x1, -2*x2, 1)         (16x4)
//   B[n] = ( y0,    y1,    y2,   |y_n|^2)   (4x16)
//   C[r] = |x_row|^2                        (loop-invariant!)
//   D    = A*B + C = |x|^2 + |y|^2 - 2*x.y
// Running min kept in 8 VGPRs (C/D layout: VGPR r = row r / r+8 per half,
// lane%16 = column). Clamp max(.,0) hoisted out of the loop (commutes with
// min), then shfl_xor min-reduce within each 16-lane half, then bitwise uint
// atomicMin (order-preserving since values are >= 0).
// Assumes N % 128 == 0 and M % (16*COL_SPLIT) == 0 (true here: 16384).
// ---------------------------------------------------------------------------
__global__ void __launch_bounds__(256)
cd_rowmin_kernel(const float* __restrict__ X, const float* __restrict__ Y,
                 unsigned* __restrict__ out_bits, int M) {
  const int lane = threadIdx.x & 31;
  const int wave = threadIdx.x >> 5;
  const int half = lane >> 4;  // 0: K=0,1 / rows r      1: K=2,3 / rows r+8
  const int l    = lane & 15;

  const int i0 = blockIdx.x * (8 * 16) + wave * 16;

  // A fragment (loaded once per wave): ISA 16x4 f32 A layout.
  //   lane m (<16)  -> (-2*x[m][0], -2*x[m][1])
  //   lane m+16     -> (-2*x[m][2], 1)
  const float* xr = X + (size_t)(i0 + l) * 3;
  const float ax0 = xr[0], ax1 = xr[1], ax2 = xr[2];
  v2f a;
  a.x = half ? (-2.0f * ax2) : (-2.0f * ax0);
  a.y = half ? 1.0f          : (-2.0f * ax1);

  // Loop-invariant accumulator: |x|^2 for the 8 rows this lane's C/D
  // registers correspond to.
  v8f c;
#pragma unroll
  for (int r = 0; r < 8; ++r) {
    const float* p = X + (size_t)(i0 + half * 8 + r) * 3;
    const float t0 = p[0], t1 = p[1], t2 = p[2];
    c[r] = t0 * t0 + t1 * t1 + t2 * t2;
  }

  float mn[8];
#pragma unroll
  for (int r = 0; r < 8; ++r) mn[r] = __builtin_inff();

  const int cols_per_seg = M / COL_SPLIT;
  const int j_begin = blockIdx.y * cols_per_seg;
  const int j_end   = j_begin + cols_per_seg;

#pragma unroll 2
  for (int j0 = j_begin; j0 < j_end; j0 += 16) {
    const int j = j0 + l;  // both halves load the same Y row -> merged in WGP$
    const float* yr = Y + (size_t)j * 3;
    const float y0 = yr[0], y1 = yr[1], y2 = yr[2];
    const float ysq = y0 * y0 + y1 * y1 + y2 * y2;

    // B fragment: ISA 4x16 f32 B layout; K=3 slot carries |y|^2.
    v2f b;
    b.x = half ? y2  : y0;
    b.y = half ? ysq : y1;

    // D = squared distance tile, straight out of the matrix unit.
    v8f d = __builtin_amdgcn_wmma_f32_16x16x4_f32(
        /*neg_a=*/false, a, /*neg_b=*/false, b,
        /*c_mod=*/(short)0, c, /*reuse_a=*/false, /*reuse_b=*/false);

#pragma unroll
    for (int r = 0; r < 8; ++r)
      mn[r] = fminf(mn[r], d[r]);
  }

  // Clamp once (min_j max(d,0) == max(min_j d, 0)); keeps values >= 0 so the
  // uint bit pattern stays order-preserving for atomicMin.
#pragma unroll
  for (int r = 0; r < 8; ++r) mn[r] = fmaxf(mn[r], 0.0f);

  // Min-reduce across the 16 lanes of each half (xor masks 8..1 stay in-half).
#pragma unroll
  for (int off = 8; off >= 1; off >>= 1) {
#pragma unroll
    for (int r = 0; r < 8; ++r)
      mn[r] = fminf(mn[r], __shfl_xor(mn[r], off, 32));
  }

  if (l == 0) {
#pragma unroll
    for (int r = 0; r < 8; ++r) {
      const int row = i0 + half * 8 + r;
      atomicMin(out_bits + row, __float_as_uint(mn[r]));
    }
  }
}

// ---------------------------------------------------------------------------
// Sum the N+M per-row/per-col minima and scale -> scalar mean.
// ---------------------------------------------------------------------------
__global__ void cd_finalize_kernel(const unsigned* __restrict__ bits, int total,
                                   float* __restrict__ out, float inv_count) {
  __shared__ float s[256];
  float acc = 0.0f;
  for (int i = threadIdx.x; i < total; i += 256)
    acc += __uint_as_float(bits[i]);
  s[threadIdx.x] = acc;
  __syncthreads();
  for (int st = 128; st > 0; st >>= 1) {
    if ((int)threadIdx.x < st) s[threadIdx.x] += s[threadIdx.x + st];
    __syncthreads();
  }
  if (threadIdx.x == 0) out[0] = s[0] * inv_count;
}

extern "C" void kernel_launch(void* const* d_in, const int* in_sizes, int n_in,
                              void* d_out, int out_size, void* d_ws, size_t ws_size,
                              hipStream_t stream) {
  const float* x = (const float*)d_in[0];
  const float* y = (const float*)d_in[1];
  const int N = in_sizes[0] / 3;
  const int M = in_sizes[1] / 3;

  unsigned* row_bits = (unsigned*)d_ws;   // N entries: min_j dist(x_i, y_j)
  unsigned* col_bits = row_bits + N;      // M entries: min_i dist(x_i, y_j)

  cd_init_kernel<<<(N + M + 255) / 256, 256, 0, stream>>>(row_bits, N + M);

  const dim3 blk(256);
  // Pass 1: row mins (x rows vs all y).
  cd_rowmin_kernel<<<dim3(N / 128, COL_SPLIT), blk, 0, stream>>>(x, y, row_bits, M);
  // Pass 2: col mins by symmetry (y rows vs all x).
  cd_rowmin_kernel<<<dim3(M / 128, COL_SPLIT), blk, 0, stream>>>(y, x, col_bits, N);

  cd_finalize_kernel<<<1, 256, 0, stream>>>(row_bits, N + M, (float*)d_out,
                                            1.0f / (float)(N + M));
}